// GraphSAGE_35639638622735
// MI455X (gfx1250) — compile-verified
//
#include <hip/hip_runtime.h>

// GraphSAGE (2-layer SAGEConv + fc residual) for MI455X / gfx1250.
// N=10000 nodes, E=640000 edges, D=128. fp32 throughout; GEMMs via
// V_WMMA_F32_16X16X4_F32 (wave32 matrix cores), edge aggregation via
// L2-resident global_atomic_add_f32 (whole working set ~41MB << 192MB L2).

#define DIM  128
#define TILE 16

typedef __attribute__((ext_vector_type(2))) float v2f;
typedef __attribute__((ext_vector_type(8))) float v8f;

// ---------------------------------------------------------------- utilities
__global__ void zero_f32(float* __restrict__ p, long n) {
    long i = (long)blockIdx.x * blockDim.x + threadIdx.x;
    const long stride = (long)gridDim.x * blockDim.x;
    for (; i < n; i += stride) p[i] = 0.0f;
}

__global__ void degree_kernel(const long long* __restrict__ dst,
                              float* __restrict__ deg, int n_edges) {
    int e = blockIdx.x * blockDim.x + threadIdx.x;
    if (e < n_edges) atomicAdd(&deg[(int)dst[e]], 1.0f);
}

// One 32-lane wave per edge; each lane moves 4 consecutive floats (float4
// gather, 4x f32 scatter-atomics). 512B coalesced per edge.
__global__ void scatter_kernel(const float* __restrict__ feat,
                               const long long* __restrict__ src,
                               const long long* __restrict__ dst,
                               float* __restrict__ agg, int n_edges) {
    unsigned idx  = blockIdx.x * blockDim.x + threadIdx.x;
    unsigned e    = idx >> 5;
    unsigned lane = idx & 31u;
    if (e >= (unsigned)n_edges) return;
    const long s = (long)src[e];
    const long d = (long)dst[e];
    const float4 v = *reinterpret_cast<const float4*>(feat + s * DIM + lane * 4);
    float* dp = agg + d * DIM + lane * 4;
    atomicAdd(dp + 0, v.x);
    atomicAdd(dp + 1, v.y);
    atomicAdd(dp + 2, v.z);
    atomicAdd(dp + 3, v.w);
}

// ---------------------------------------------------------------- fused GEMM
// out[m, :] = act( [ (agg[m,:]/max(deg,1)) @ LW^T ] + X[m,:] @ RW^T + lb + res[m,:] )
// Block = 256 threads = 8 waves. Block handles a 16-node row tile (staged in
// LDS); each wave owns one 16-column tile and runs K=128 in 4-wide WMMA steps.
// Requires n_nodes % 16 == 0 (10000 = 625*16) and full EXEC (no divergence).
template <bool HAS_AGG, bool HAS_RES, bool RELU>
__global__ __launch_bounds__(256) void sage_gemm(
    const float* __restrict__ agg, const float* __restrict__ deg,
    const float* __restrict__ X,
    const float* __restrict__ LW, const float* __restrict__ lb,
    const float* __restrict__ RW,
    const float* __restrict__ res, float* __restrict__ out) {
    __shared__ float sA[HAS_AGG ? TILE * DIM : 1];
    __shared__ float sX[TILE * DIM];

    const int  tid  = threadIdx.x;
    const long base = (long)blockIdx.x * TILE;  // first node row of this tile

    for (int i = tid; i < TILE * DIM; i += 256) {
        const int  r    = i >> 7;       // row within tile
        const int  c    = i & 127;      // feature column
        const long node = base + r;
        sX[i] = X[node * DIM + c];
        if (HAS_AGG) {
            const float inv = 1.0f / fmaxf(deg[node], 1.0f);
            sA[i] = agg[node * DIM + c] * inv;
        }
    }
    __syncthreads();

    const int lane = tid & 31;
    const int n0   = (tid >> 5) * TILE;   // wave's column-tile origin
    const int m    = lane & 15;           // A row (and B column) for this lane
    const int kh   = (lane >> 4) << 1;    // K sub-offset: lanes 0-15 -> 0, 16-31 -> 2

    v8f acc = {};

    if (HAS_AGG) {
#pragma unroll 8
        for (int k0 = 0; k0 < DIM; k0 += 4) {
            v2f a, b;
            a.x = sA[m * DIM + k0 + kh];
            a.y = sA[m * DIM + k0 + kh + 1];
            const float* bp = LW + (long)(n0 + m) * DIM + k0 + kh;  // (LW^T)[k][n]
            b.x = bp[0];
            b.y = bp[1];
            acc = __builtin_amdgcn_wmma_f32_16x16x4_f32(
                false, a, false, b, (short)0, acc, false, false);
        }
    }
#pragma unroll 8
    for (int k0 = 0; k0 < DIM; k0 += 4) {
        v2f a, b;
        a.x = sX[m * DIM + k0 + kh];
        a.y = sX[m * DIM + k0 + kh + 1];
        const float* bp = RW + (long)(n0 + m) * DIM + k0 + kh;      // (RW^T)[k][n]
        b.x = bp[0];
        b.y = bp[1];
        acc = __builtin_amdgcn_wmma_f32_16x16x4_f32(
            false, a, false, b, (short)0, acc, false, false);
    }

    // Epilogue. C/D layout: VGPR r, lanes 0-15 -> M=r, lanes 16-31 -> M=r+8.
    const int   n    = n0 + m;
    const float bias = lb[n];
    const int   mb   = (lane >> 4) * 8;
#pragma unroll
    for (int r = 0; r < 8; ++r) {
        const long node = base + mb + r;
        float v = acc[r] + bias;
        if (HAS_RES) v += res[node * DIM + n];
        if (RELU) v = fmaxf(v, 0.0f);
        out[node * DIM + n] = v;
    }
}

// ---------------------------------------------------------------- launcher
extern "C" void kernel_launch(void* const* d_in, const int* in_sizes, int n_in,
                              void* d_out, int out_size, void* d_ws, size_t ws_size,
                              hipStream_t stream) {
    const float*     x    = (const float*)d_in[0];
    const long long* ei   = (const long long*)d_in[1];
    const float*     fc_w = (const float*)d_in[2];
    const float*     fc_b = (const float*)d_in[3];
    const float*     f_lw = (const float*)d_in[4];
    const float*     f_lb = (const float*)d_in[5];
    const float*     f_rw = (const float*)d_in[6];
    const float*     n_lw = (const float*)d_in[7];
    const float*     n_lb = (const float*)d_in[8];
    const float*     n_rw = (const float*)d_in[9];
    float*           out  = (float*)d_out;

    const int  n_nodes = in_sizes[0] / DIM;   // 10000 (multiple of 16)
    const int  n_edges = in_sizes[1] / 2;     // 640000
    const long nd      = (long)n_nodes * DIM;

    // workspace carve-up (float-aligned): deg | agg | h | inp
    float* ws  = (float*)d_ws;
    float* deg = ws;
    float* agg = ws + ((n_nodes + 63) & ~63);
    float* h   = agg + nd;
    float* inp = h + nd;

    const long long* src = ei;             // edge_index[0]
    const long long* dst = ei + n_edges;   // edge_index[1]

    const int  eb = (n_edges + 255) / 256;
    const int  sb = (int)(((long)n_edges * 32 + 255) / 256);
    const int  gb = n_nodes / TILE;        // 625 row tiles

    // zero deg..agg+nd (contiguous span, covers alignment pad)
    zero_f32<<<1024, 256, 0, stream>>>(deg, (long)(agg - deg) + nd);

    // degrees (shared by both layers)
    degree_kernel<<<eb, 256, 0, stream>>>(dst, deg, n_edges);

    // ---- layer 1 aggregation: agg = sum_{e:dst=v} x[src(e)]
    scatter_kernel<<<sb, 256, 0, stream>>>(x, src, dst, agg, n_edges);

    // fc residual branch: inp = x @ fc_w^T + fc_b
    sage_gemm<false, false, false><<<gb, 256, 0, stream>>>(
        nullptr, nullptr, x, nullptr, fc_b, fc_w, nullptr, inp);

    // h = relu( (agg/deg) @ f_lw^T + f_lb + x @ f_rw^T )
    sage_gemm<true, false, true><<<gb, 256, 0, stream>>>(
        agg, deg, x, f_lw, f_lb, f_rw, nullptr, h);

    // ---- layer 2 aggregation on h
    zero_f32<<<1024, 256, 0, stream>>>(agg, nd);
    scatter_kernel<<<sb, 256, 0, stream>>>(h, src, dst, agg, n_edges);

    // out = relu( (agg/deg) @ n_lw^T + n_lb + h @ n_rw^T + inp )
    sage_gemm<true, true, true><<<gb, 256, 0, stream>>>(
        agg, deg, h, n_lw, n_lb, n_rw, inp, out);
}